// End2End_74689481277987
// MI455X (gfx1250) — compile-verified
//
#include <hip/hip_runtime.h>
#include <cstdint>
#include <cstddef>

// ---------------------------------------------------------------------------
// Top-K detection post-processing for MI455X (gfx1250, wave32).
// One workgroup per batch image. Scores are DMA'd into LDS once via the
// Tensor Data Mover (TENSORcnt), then a 3-level radix select + bitonic sort
// runs entirely out of LDS (b128 sweeps). Boxes are gathered from HBM only
// at the 1000 winning positions.
// ---------------------------------------------------------------------------

typedef __attribute__((ext_vector_type(4))) unsigned int v4u;
typedef __attribute__((ext_vector_type(8))) int          v8i;
typedef __attribute__((ext_vector_type(4))) int          v4i;

#define NT       1024         // threads per block (32 waves of 32 = WG max)
#define N0       25600        // 160*160, stride 8
#define N1       6400         // 80*80,   stride 16
#define N2       1600         // 40*40,   stride 32
#define N_TOTAL  33600
#define NV4      (N_TOTAL / 4)   // 8400 uint4 groups
#define KTOP     1000
#define CAP      2048         // candidate buffer (power of 2 >= KTOP + expected ties)
#define BATCH    128

// LDS carve (dynamic shared memory):
//   [0,          134400)  sKey   : 33600 u32 (float bits -> sortable keys in place)
//   [134400,     150784)  sHist  : 4096 u32 histogram
//   [150784,     167168)  sCand  : 2048 u64 candidates (key<<32 | ~idx)
//   [167168,     168192)  sChunk : 256 u32 partial sums
//   [168192,     168224)  sRes   : 8 u32 broadcast slots
#define SMEM_BYTES 168224

// --- TDM: 1D row load, Global -> LDS, data_size = 4 bytes -------------------
// D# layout per CDNA5 ISA ch. 7/8 (CDNA5 ISA).
__device__ __forceinline__ void tdm_load_row_f32(uint32_t lds_byte_addr,
                                                 const float* gsrc,
                                                 uint32_t nelem) {
  uint64_t ga = (uint64_t)(uintptr_t)gsrc;
  v4u g0;
  g0[0] = 1u;                                                 // count=1, user descriptor
  g0[1] = lds_byte_addr;                                      // lds_addr (bytes)
  g0[2] = (uint32_t)ga;                                       // global_addr[31:0]
  g0[3] = (uint32_t)((ga >> 32) & 0x01FFFFFFu) | (2u << 30);  // global_addr[56:32] | type=2

  uint32_t w[8] = {0u, 0u, 0u, 0u, 0u, 0u, 0u, 0u};
  w[0]  = (2u << 16);                    // data_size = 2 (4 bytes); no multicast/pad/iterate
  w[1] |= (nelem & 0xFFFFu) << 16;       // tensor_dim0[15:0]   (bits 63:48)
  w[2] |= (nelem >> 16) & 0xFFFFu;       // tensor_dim0[31:16]  (bits 79:64)
  w[2] |= (1u << 16);                    // tensor_dim1 = 1     (bits 95:80)
  w[3] |= (nelem & 0xFFFFu) << 16;       // tile_dim0           (bits 127:112)
  w[4]  = 1u;                            // tile_dim1 = 1       (bits 143:128)
  w[5]  = nelem;                         // tensor_dim0_stride[31:0] (bits 191:160)

  v8i g1;
#pragma unroll
  for (int i = 0; i < 8; ++i) g1[i] = (int)w[i];
  v4i z4 = {0, 0, 0, 0};
  v8i z8 = {0, 0, 0, 0, 0, 0, 0, 0};
  // 6-arg form (clang-23 / therock-10.0 headers): (g0, g1, g2, g3, extra, cpol)
  __builtin_amdgcn_tensor_load_to_lds(g0, g1, z4, z4, z8, 0);
}

__device__ __forceinline__ uint32_t to_key(uint32_t fb) {
  return (fb & 0x80000000u) ? ~fb : (fb | 0x80000000u);
}

// Find (from top) bucket T s.t. sum_{j>T} hist[j] < need <= sum_{j>=T}.
// sRes[0] = T, sRes[1] = count strictly above T. H is 4096 or 256.
__device__ __forceinline__ void find_bucket(uint32_t* hist, int H, uint32_t need,
                                            uint32_t* sChunk, uint32_t* sRes, int tid) {
  const int cs = H >> 8;  // entries per chunk, 256 chunks
  if (tid < 256) {
    uint32_t s = 0;
    for (int j = 0; j < cs; ++j) s += hist[tid * cs + j];
    sChunk[tid] = s;
  }
  __syncthreads();
  if (tid == 0) {
    uint32_t run = 0;
    int c = 255;
    for (; c > 0; --c) {
      uint32_t v = sChunk[c];
      if (run + v >= need) break;
      run += v;
    }
    int T = c * cs;
    for (int j = cs - 1; j >= 0; --j) {
      uint32_t v = hist[c * cs + j];
      if (run + v >= need) { T = c * cs + j; break; }
      run += v;
    }
    sRes[0] = (uint32_t)T;
    sRes[1] = run;
  }
  __syncthreads();
}

__global__ __launch_bounds__(NT) void topk_detect_kernel(
    const float* __restrict__ cls0, const float* __restrict__ cls1,
    const float* __restrict__ cls2, const float* __restrict__ bb0,
    const float* __restrict__ bb1,  const float* __restrict__ bb2,
    float* __restrict__ out) {
  extern __shared__ unsigned char smem[];
  uint32_t* sKey   = (uint32_t*)(smem);
  uint4*    sKey4  = (uint4*)(smem);
  uint32_t* sHist  = (uint32_t*)(smem + 134400);
  uint64_t* sCand  = (uint64_t*)(smem + 150784);
  uint32_t* sChunk = (uint32_t*)(smem + 167168);
  uint32_t* sRes   = (uint32_t*)(smem + 168192);

  const int b   = blockIdx.x;
  const int tid = threadIdx.x;
  // Low 32 bits of the generic pointer to LDS == byte offset within the
  // workgroup's LDS allocation.
  const uint32_t ldsBase = (uint32_t)(uintptr_t)smem;

  // ---- Phase 0: TDM the whole batch's score map into LDS (wave 0 only;
  // TDM issue is per-wave and ignores EXEC). ------------------------------
  if (tid < 32) {
    tdm_load_row_f32(ldsBase,                  cls0 + (size_t)b * N0, N0);
    tdm_load_row_f32(ldsBase + N0 * 4u,        cls1 + (size_t)b * N1, N1);
    tdm_load_row_f32(ldsBase + (N0 + N1) * 4u, cls2 + (size_t)b * N2, N2);
    __builtin_amdgcn_s_wait_tensorcnt(0);
  }
  __syncthreads();

  // ---- Phase 1: sortable-key conversion in place + 12-bit histogram -----
  for (int i = tid; i < 4096; i += NT) sHist[i] = 0u;
  __syncthreads();
  for (int i = tid; i < NV4; i += NT) {
    uint4 v = sKey4[i];
    v.x = to_key(v.x); v.y = to_key(v.y); v.z = to_key(v.z); v.w = to_key(v.w);
    sKey4[i] = v;
    atomicAdd(&sHist[v.x >> 20], 1u);
    atomicAdd(&sHist[v.y >> 20], 1u);
    atomicAdd(&sHist[v.z >> 20], 1u);
    atomicAdd(&sHist[v.w >> 20], 1u);
  }
  __syncthreads();
  find_bucket(sHist, 4096, KTOP, sChunk, sRes, tid);
  const uint32_t T1 = sRes[0];
  const uint32_t need2 = KTOP - sRes[1];
  __syncthreads();

  // ---- Phase 2: refine middle 12 bits within bucket T1 ------------------
  for (int i = tid; i < 4096; i += NT) sHist[i] = 0u;
  __syncthreads();
  for (int i = tid; i < NV4; i += NT) {
    uint4 v = sKey4[i];
    if ((v.x >> 20) == T1) atomicAdd(&sHist[(v.x >> 8) & 0xFFFu], 1u);
    if ((v.y >> 20) == T1) atomicAdd(&sHist[(v.y >> 8) & 0xFFFu], 1u);
    if ((v.z >> 20) == T1) atomicAdd(&sHist[(v.z >> 8) & 0xFFFu], 1u);
    if ((v.w >> 20) == T1) atomicAdd(&sHist[(v.w >> 8) & 0xFFFu], 1u);
  }
  __syncthreads();
  find_bucket(sHist, 4096, need2, sChunk, sRes, tid);
  const uint32_t T2 = sRes[0];
  const uint32_t need3 = need2 - sRes[1];
  const uint32_t pref24 = (T1 << 12) | T2;
  __syncthreads();

  // ---- Phase 3: refine last 8 bits -> exact threshold key ---------------
  for (int i = tid; i < 256; i += NT) sHist[i] = 0u;
  __syncthreads();
  for (int i = tid; i < NV4; i += NT) {
    uint4 v = sKey4[i];
    if ((v.x >> 8) == pref24) atomicAdd(&sHist[v.x & 0xFFu], 1u);
    if ((v.y >> 8) == pref24) atomicAdd(&sHist[v.y & 0xFFu], 1u);
    if ((v.z >> 8) == pref24) atomicAdd(&sHist[v.z & 0xFFu], 1u);
    if ((v.w >> 8) == pref24) atomicAdd(&sHist[v.w & 0xFFu], 1u);
  }
  __syncthreads();
  find_bucket(sHist, 256, need3, sChunk, sRes, tid);
  const uint32_t Kth = (pref24 << 8) | sRes[0];
  if (tid == 0) sRes[4] = 0u;
  __syncthreads();

  // ---- Phase 4: collect candidates (key >= Kth), pack (key desc, idx asc)
  for (int i = tid; i < NV4; i += NT) {
    uint4 v = sKey4[i];
    uint32_t base = (uint32_t)i * 4u;
#pragma unroll
    for (int c = 0; c < 4; ++c) {
      uint32_t key = (c == 0) ? v.x : (c == 1) ? v.y : (c == 2) ? v.z : v.w;
      if (key >= Kth) {
        uint32_t pos = atomicAdd(&sRes[4], 1u);
        if (pos < CAP)
          sCand[pos] = ((uint64_t)key << 32) | (uint64_t)(~(base + (uint32_t)c));
      }
    }
  }
  __syncthreads();
  uint32_t total = sRes[4];
  if (total > CAP) total = CAP;
  for (int i = tid; i < CAP; i += NT)
    if ((uint32_t)i >= total) sCand[i] = 0ull;  // -inf-like pad, sinks to bottom
  __syncthreads();

  // ---- Phase 5: bitonic sort CAP composites, descending -----------------
  for (unsigned k = 2; k <= CAP; k <<= 1) {
    for (unsigned j = k >> 1; j > 0; j >>= 1) {
      for (unsigned t = tid; t < CAP; t += NT) {
        unsigned ixj = t ^ j;
        if (ixj > t) {
          uint64_t a = sCand[t], c = sCand[ixj];
          bool desc = ((t & k) == 0);
          if (desc ? (a < c) : (a > c)) { sCand[t] = c; sCand[ixj] = a; }
        }
      }
      __syncthreads();
    }
  }

  // ---- Phase 6: emit sorted top-1000 scores + decoded boxes -------------
  float* outBoxes  = out;                               // (128,1000,4)
  float* outScores = out + (size_t)BATCH * KTOP * 4;    // (128,1000,1)
  for (int i = tid; i < KTOP; i += NT) {
    uint64_t c   = sCand[i];
    uint32_t key = (uint32_t)(c >> 32);
    uint32_t idx = ~((uint32_t)c);
    uint32_t fb  = (key & 0x80000000u) ? (key & 0x7FFFFFFFu) : ~key;
    float logit  = __uint_as_float(fb);
    float score  = 1.0f / (1.0f + expf(-logit));

    const float* bb;
    int pos, HW, W;
    float strd;
    if (idx < N0) {
      bb = bb0 + (size_t)b * 4 * N0; pos = (int)idx;           HW = N0; W = 160; strd = 8.0f;
    } else if (idx < N0 + N1) {
      bb = bb1 + (size_t)b * 4 * N1; pos = (int)idx - N0;      HW = N1; W = 80;  strd = 16.0f;
    } else {
      bb = bb2 + (size_t)b * 4 * N2; pos = (int)idx - N0 - N1; HW = N2; W = 40;  strd = 32.0f;
    }
    int x = pos % W, y = pos / W;
    float px = (float)x * strd;
    float py = (float)y * strd;
    float l = bb[pos];
    float t = bb[HW + pos];
    float r = bb[2 * HW + pos];
    float d = bb[3 * HW + pos];

    float4 box;
    box.x = px - l; box.y = py - t; box.z = px + r; box.w = py + d;
    *(float4*)(outBoxes + ((size_t)b * KTOP + (size_t)i) * 4) = box;
    outScores[(size_t)b * KTOP + (size_t)i] = score;
  }
}

extern "C" void kernel_launch(void* const* d_in, const int* in_sizes, int n_in,
                              void* d_out, int out_size, void* d_ws, size_t ws_size,
                              hipStream_t stream) {
  // setup_inputs() insertion order: cls0, bbox0, cls1, bbox1, cls2, bbox2
  const float* cls0 = (const float*)d_in[0];
  const float* bb0  = (const float*)d_in[1];
  const float* cls1 = (const float*)d_in[2];
  const float* bb1  = (const float*)d_in[3];
  const float* cls2 = (const float*)d_in[4];
  const float* bb2  = (const float*)d_in[5];
  float* out = (float*)d_out;

  // Allow >64KB dynamic LDS (CDNA5: up to 320KB per workgroup). Idempotent.
  (void)hipFuncSetAttribute((const void*)topk_detect_kernel,
                            hipFuncAttributeMaxDynamicSharedMemorySize,
                            (int)SMEM_BYTES);

  topk_detect_kernel<<<BATCH, NT, SMEM_BYTES, stream>>>(cls0, cls1, cls2,
                                                        bb0, bb1, bb2, out);
}